// GlobalGraphAttentionLayer_42958262895020
// MI455X (gfx1250) — compile-verified
//
#include <hip/hip_runtime.h>
#include <math.h>

// Sizes fixed by the reference
#define BB 32
#define NN 90
#define EE 8100            // NN*NN
#define ME ((size_t)BB * EE)

// K-split GEMM geometry
#define KSPLIT 5
#define SPLITLEN 1620      // EE / KSPLIT
#define KC 324             // X columns staged per LDS chunk
#define NCHUNK 5           // SPLITLEN / KC
#define XSTRIDE 332        // padded LDS row stride in dwords (16B aligned, conflict-free)
#define GRP 9              // k-steps per pipeline group (81 steps = 9 groups of 9)

typedef float v2f __attribute__((ext_vector_type(2)));
typedef float v8f __attribute__((ext_vector_type(8)));

struct P3 { const float* p[3]; };

// ---------------------------------------------------------------------------
// Partial GEMM: P[batch][split][32, Ncols] = X[32, Ksub] @ W[Ncols, Ksub]^T
// over k in [split*SPLITLEN, (split+1)*SPLITLEN).  One wave per 16-col tile,
// two 16-row M tiles share each B tile (V_WMMA_F32_16X16X4_F32).
// X chunks staged to LDS via gfx1250 async-to-LDS loads (ASYNCcnt);
// W streamed from global with a double-buffered register pipeline + prefetch.
// ---------------------------------------------------------------------------
__global__ __launch_bounds__(256) void gemm32_split(
    P3 Xs, P3 Ws, float* __restrict__ P, int K, int Ncols)
{
    __shared__ float sX[32 * XSTRIDE];   // 42.5 KB

    const int tid   = threadIdx.x;
    const int lane  = tid & 31;
    const int wave  = tid >> 5;
    const int batch = blockIdx.y;
    const int split = blockIdx.z;

    const float* X  = Xs.p[batch];
    const float* Wt = Ws.p[batch];

    const int  jt     = blockIdx.x * 8 + wave;
    const int  jb     = jt * 16;
    const bool active = (jb < Ncols);          // wave-uniform

    const int nsub  = lane & 15;               // output column within tile
    const int khalf = lane >> 4;               // k-pair select (0/1)
    const int j     = jb + nsub;
    const int jc    = active ? ((j < Ncols) ? j : (Ncols - 1)) : 0;

    const float* wrow = Wt + (size_t)jc * K + 2 * khalf;

    v8f c0 = {};
    v8f c1 = {};
    const int kbase = split * SPLITLEN;

    for (int ch = 0; ch < NCHUNK; ++ch) {
        const int k0 = kbase + ch * KC;

        __syncthreads();   // previous chunk fully consumed
        // Stage X[0:32, k0:k0+KC] into LDS: 32*81 b128 lane-transfers (ASYNCcnt).
        for (int idx = tid; idx < 32 * (KC / 4); idx += 256) {
            const int row = idx / (KC / 4);
            const int c16 = idx - row * (KC / 4);
            unsigned lds = (unsigned)(size_t)(&sX[row * XSTRIDE + c16 * 4]);
            unsigned long long ga =
                (unsigned long long)(size_t)(X + (size_t)row * K + k0 + c16 * 4);
            asm volatile("global_load_async_to_lds_b128 %0, %1, off"
                         :: "v"(lds), "v"(ga) : "memory");
        }
        asm volatile("s_wait_asynccnt 0x0" ::: "memory");
        __syncthreads();   // chunk visible to all waves

        if (active) {
            const float* a0p   = &sX[nsub * XSTRIDE + 2 * khalf];
            const float* a1p   = &sX[(16 + nsub) * XSTRIDE + 2 * khalf];
            const float* wbase = wrow + k0;

            // Double-buffered W register pipeline: GRP loads in flight
            // across every WMMA burst.
            v2f wb[2][GRP];
            #pragma unroll
            for (int u = 0; u < GRP; ++u)
                wb[0][u] = *(const v2f*)(wbase + 4 * u);

            #pragma unroll
            for (int g = 0; g < GRP; ++g) {
                const int kg = g * (4 * GRP);
                if (g + 1 < GRP) {
                    #pragma unroll
                    for (int u = 0; u < GRP; ++u)
                        wb[(g + 1) & 1][u] =
                            *(const v2f*)(wbase + kg + 4 * GRP + 4 * u);
                }
                __builtin_prefetch(wbase + kg + 1024, 0, 3);

                v2f a0[GRP], a1[GRP];
                #pragma unroll
                for (int u = 0; u < GRP; ++u) {
                    a0[u] = *(const v2f*)(a0p + kg + 4 * u);   // ds_load_b64
                    a1[u] = *(const v2f*)(a1p + kg + 4 * u);
                }
                #pragma unroll
                for (int u = 0; u < GRP; ++u) {
                    c0 = __builtin_amdgcn_wmma_f32_16x16x4_f32(
                             false, a0[u], false, wb[g & 1][u], (short)0, c0, false, false);
                    c1 = __builtin_amdgcn_wmma_f32_16x16x4_f32(
                             false, a1[u], false, wb[g & 1][u], (short)0, c1, false, false);
                }
            }
        }
    }

    if (active && j < Ncols) {
        float* Pb = P + ((size_t)batch * KSPLIT + split) * 32 * (size_t)Ncols;
        #pragma unroll
        for (int r = 0; r < 8; ++r) {
            const int m = r + 8 * khalf;       // D layout: VGPR r -> row r + 8*(lane>>4)
            Pb[(size_t)m * Ncols + j]        = c0[r];
            Pb[(size_t)(16 + m) * Ncols + j] = c1[r];
        }
    }
}

// ---------------------------------------------------------------------------
// Deterministic reduction over K-splits + bias (+ optional tanh).
// C is contiguous [nb][32][Ncols].
// ---------------------------------------------------------------------------
__global__ __launch_bounds__(256) void reduce_bias_kernel(
    const float* __restrict__ P, P3 Bs, float* __restrict__ C,
    int Ncols, int nb, int fuse_tanh)
{
    const size_t total = (size_t)nb * 32 * Ncols;
    const size_t idx = (size_t)blockIdx.x * 256 + threadIdx.x;
    if (idx >= total) return;
    const size_t per = (size_t)32 * Ncols;
    const int    b   = (int)(idx / per);
    const size_t rem = idx - (size_t)b * per;
    const int    j   = (int)(rem % Ncols);
    float s = 0.f;
    #pragma unroll
    for (int ss = 0; ss < KSPLIT; ++ss)
        s += P[((size_t)b * KSPLIT + ss) * per + rem];
    s += Bs.p[b][j];
    if (fuse_tanh) s = tanhf(s);
    C[idx] = s;
}

// ---------------------------------------------------------------------------
// Per-head attention: one wave per head (wave32), lane = row i.
// ---------------------------------------------------------------------------
__global__ __launch_bounds__(256) void attn_heads(
    const float* __restrict__ q2, const float* __restrict__ k2,
    const float* __restrict__ v2, float* __restrict__ o)
{
    const int h = blockIdx.x * 8 + (threadIdx.x >> 5);
    if (h >= EE) return;
    const int i = threadIdx.x & 31;

    const float qi = q2[(size_t)i * EE + h];
    const float ki = k2[(size_t)i * EE + h];
    const float vi = v2[(size_t)i * EE + h];

    float m = -INFINITY;
    #pragma unroll
    for (int j = 0; j < 32; ++j) {
        float kj = __shfl(ki, j, 32);
        m = fmaxf(m, qi * kj);
    }
    float sum = 0.f, acc = 0.f;
    #pragma unroll
    for (int j = 0; j < 32; ++j) {
        float kj = __shfl(ki, j, 32);
        float vj = __shfl(vi, j, 32);
        float e  = expf(qi * kj - m);
        sum += e;
        acc += e * vj;
    }
    o[(size_t)i * EE + h] = acc / sum;
}

// ---------------------------------------------------------------------------
// Wh[b] = h[b] @ W   (90x90 per batch), W staged in LDS.
// ---------------------------------------------------------------------------
__global__ __launch_bounds__(256) void wh_kernel(
    const float* __restrict__ h, const float* __restrict__ W,
    float* __restrict__ Wh)
{
    __shared__ float sW[NN * NN];
    const int b = blockIdx.x;
    const float* hb = h + (size_t)b * EE;
    for (int idx = threadIdx.x; idx < EE; idx += 256) sW[idx] = W[idx];
    __syncthreads();
    float* outb = Wh + (size_t)b * EE;
    for (int idx = threadIdx.x; idx < EE; idx += 256) {
        const int i = idx / NN, jj = idx % NN;
        float acc = 0.f;
        for (int l = 0; l < NN; ++l) acc += hb[i * NN + l] * sW[l * NN + jj];
        outb[idx] = acc;
    }
}

// ---------------------------------------------------------------------------
// Batched PageRank power iteration (100 iters).
// ---------------------------------------------------------------------------
__global__ __launch_bounds__(96) void pagerank_kernel(
    const float* __restrict__ adj, float* __restrict__ pr)
{
    __shared__ float S[NN * 91];
    __shared__ float xbuf[2][NN];
    __shared__ float dang[NN];
    const int b = blockIdx.x;
    const int t = threadIdx.x;
    const float* A = adj + (size_t)b * EE;

    for (int idx = t; idx < EE; idx += 96) {
        const int i = idx / NN, j = idx % NN;
        S[i * 91 + j] = (i == j) ? 0.f : A[idx];
    }
    __syncthreads();
    if (t < NN) {
        float s = 0.f;
        for (int j = 0; j < NN; ++j) s += S[t * 91 + j];
        const float inv = (s > 0.f) ? (1.f / s) : 0.f;
        dang[t] = (s > 0.f) ? 0.f : 1.f;
        for (int j = 0; j < NN; ++j) S[t * 91 + j] *= inv;
        xbuf[0][t] = 1.f / (float)NN;
    }
    __syncthreads();

    const float damping = 0.85f;
    int cur = 0;
    for (int it = 0; it < 100; ++it) {
        if (t < NN) {
            float d = 0.f, acc = 0.f;
            for (int i2 = 0; i2 < NN; ++i2) {
                const float xi = xbuf[cur][i2];
                d   += xi * dang[i2];
                acc += xi * S[i2 * 91 + t];
            }
            xbuf[1 - cur][t] = damping * (acc + d * (1.f / NN))
                             + (1.f - damping) * (1.f / NN);
        }
        __syncthreads();
        cur = 1 - cur;
    }
    if (t < NN) pr[b * NN + t] = xbuf[cur][t];
}

// ---------------------------------------------------------------------------
// edge_att row-softmax + global_att = node_att + edge_att + final matmul.
// ---------------------------------------------------------------------------
__global__ __launch_bounds__(128) void final_kernel(
    const float* __restrict__ node_att,
    const float* __restrict__ adj,
    const float* __restrict__ pr,
    const float* __restrict__ Wh,
    float* __restrict__ out)
{
    __shared__ float sG[NN * NN];
    const int b = blockIdx.x;
    const int t = threadIdx.x;
    const float* ab  = adj      + (size_t)b * EE;
    const float* nb  = node_att + (size_t)b * EE;
    const float* whb = Wh       + (size_t)b * EE;

    if (t < NN) {
        const int i = t;
        const float pri = pr[b * NN + i];
        float m = -9e15f;
        for (int j = 0; j < NN; ++j)
            m = fmaxf(m, (ab[i * NN + j] > 0.f) ? pri : -9e15f);
        float sum = 0.f;
        for (int j = 0; j < NN; ++j) {
            const float mv = (ab[i * NN + j] > 0.f) ? pri : -9e15f;
            sum += expf(mv - m);
        }
        for (int j = 0; j < NN; ++j) {
            const float mv = (ab[i * NN + j] > 0.f) ? pri : -9e15f;
            sG[i * NN + j] = nb[i * NN + j] + expf(mv - m) / sum;
        }
    }
    __syncthreads();

    for (int idx = t; idx < EE; idx += 128) {
        const int i = idx / NN, l = idx % NN;
        float acc = 0.f;
        for (int j = 0; j < NN; ++j) acc += sG[i * NN + j] * whb[j * NN + l];
        out[(size_t)b * EE + idx] = acc;
    }
}

// ---------------------------------------------------------------------------
extern "C" void kernel_launch(void* const* d_in, const int* in_sizes, int n_in,
                              void* d_out, int out_size, void* d_ws, size_t ws_size,
                              hipStream_t stream) {
    const float* h     = (const float*)d_in[0];
    const float* adj   = (const float*)d_in[1];
    const float* W     = (const float*)d_in[2];
    const float* q_w   = (const float*)d_in[3];
    const float* q_b   = (const float*)d_in[4];
    const float* k_w   = (const float*)d_in[5];
    const float* k_b   = (const float*)d_in[6];
    const float* v_w   = (const float*)d_in[7];
    const float* v_b   = (const float*)d_in[8];
    const float* in_w  = (const float*)d_in[9];
    const float* in_b  = (const float*)d_in[10];
    const float* out_w = (const float*)d_in[11];
    const float* out_b = (const float*)d_in[12];
    float* out = (float*)d_out;

    float* ws   = (float*)d_ws;
    float* q    = ws + 0 * ME;          // q,k,v contiguous -> stage-2 inputs
    float* k    = ws + 1 * ME;
    float* v    = ws + 2 * ME;
    float* q2   = ws + 3 * ME;          // q2,k2,v2 contiguous
    float* o    = ws + 6 * ME;
    float* natt = ws + 7 * ME;
    float* Wh   = ws + 8 * ME;
    float* pr   = ws + 9 * ME;
    float* P    = ws + 10 * ME;         // partials: up to [3][KSPLIT][32][EE]

    const size_t EE2 = (size_t)EE * EE;
    const int gemmBlocksX = ((EE + 15) / 16 + 7) / 8;    // 64

    // ---- stage 1: q/k/v = h @ {q_w,k_w,v_w}^T + bias (fused, K-split) ----
    {
        P3 Xs = {{h, h, h}};
        P3 Ws = {{q_w, k_w, v_w}};
        gemm32_split<<<dim3(gemmBlocksX, 3, KSPLIT), 256, 0, stream>>>(Xs, Ws, P, EE, EE);
        P3 Bs = {{q_b, k_b, v_b}};
        const size_t total = (size_t)3 * 32 * EE;
        reduce_bias_kernel<<<(unsigned)((total + 255) / 256), 256, 0, stream>>>(P, Bs, q, EE, 3, 0);
    }

    // ---- stage 2: q2/k2/v2 = {q,k,v} @ {wq,wk,wv}^T + bias ----
    {
        P3 Xs = {{q, k, v}};
        P3 Ws = {{in_w, in_w + EE2, in_w + 2 * EE2}};
        gemm32_split<<<dim3(gemmBlocksX, 3, KSPLIT), 256, 0, stream>>>(Xs, Ws, P, EE, EE);
        P3 Bs = {{in_b, in_b + EE, in_b + 2 * EE}};
        const size_t total = (size_t)3 * 32 * EE;
        reduce_bias_kernel<<<(unsigned)((total + 255) / 256), 256, 0, stream>>>(P, Bs, q2, EE, 3, 0);
    }

    // ---- per-head attention ----
    attn_heads<<<(EE + 7) / 8, 256, 0, stream>>>(q2, q2 + ME, q2 + 2 * ME, o);

    // ---- output projection: node_att = tanh(o @ out_w^T + out_b) ----
    {
        P3 Xs = {{o, o, o}};
        P3 Ws = {{out_w, out_w, out_w}};
        gemm32_split<<<dim3(gemmBlocksX, 1, KSPLIT), 256, 0, stream>>>(Xs, Ws, P, EE, EE);
        P3 Bs = {{out_b, out_b, out_b}};
        const size_t total = (size_t)32 * EE;
        reduce_bias_kernel<<<(unsigned)((total + 255) / 256), 256, 0, stream>>>(P, Bs, natt, EE, 1, 1);
    }

    // ---- edge path ----
    wh_kernel<<<BB, 256, 0, stream>>>(h, W, Wh);
    pagerank_kernel<<<BB, 96, 0, stream>>>(adj, pr);
    final_kernel<<<BB, 128, 0, stream>>>(natt, adj, pr, Wh, out);
}